// StereoRegularizationLoss_38268158607530
// MI455X (gfx1250) — compile-verified
//
#include <hip/hip_runtime.h>
#include <stdint.h>

// ---------------------------------------------------------------------------
// StereoRegularizationLoss for MI455X (gfx1250, wave32)
//
// Fused single pass: warp + Charbonnier + 3x3 SSIM + LR-consistency + reduce.
// CDNA5-specific: async global->LDS B128 staging (ASYNCcnt) for the left-image
// and disparity halo tiles, s_wait_asynccnt synchronization.
// Bilinear gathers fetch sample pairs as single unaligned b64 loads.
// ---------------------------------------------------------------------------

namespace {
constexpr int kH = 512;
constexpr int kW = 960;
constexpr int kC = 3;
constexpr int kHW = kH * kW;

constexpr int TILE_H = 16;
constexpr int TILE_W = 64;
constexpr int TILES_X = kW / TILE_W;  // 15
constexpr int TILES_Y = kH / TILE_H;  // 32
constexpr int HALO_H = TILE_H + 2;    // 18
constexpr int HALO_W = TILE_W + 2;    // 66
constexpr int NCHUNK = 18;            // 18 x float4 chunks per LDS row
constexpr int ROWF = NCHUNK * 4;      // 72 floats per LDS row (16B aligned)
constexpr int NT = 256;               // 8 waves of 32

constexpr float kC1 = 0.01f * 0.01f;
constexpr float kC2 = 0.03f * 0.03f;
constexpr float kEps2 = 0.001f * 0.001f;  // Charbonnier eps^2
constexpr float kAlpha = 0.85f;
constexpr float kWlr = 0.2f;
}  // namespace

// Async copy 16B global -> LDS, tracked by ASYNCcnt (gfx1250).
__device__ __forceinline__ void async_load_b128(uint32_t lds_byte_addr,
                                                const float* gaddr) {
  asm volatile("global_load_async_to_lds_b128 %0, %1, off"
               :: "v"(lds_byte_addr), "v"(gaddr)
               : "memory");
}

__device__ __forceinline__ void wait_async_all() {
  asm volatile("s_wait_asynccnt 0x0" ::: "memory");
}

// Low 32 bits of a flat LDS address are the workgroup-relative LDS byte offset.
__device__ __forceinline__ uint32_t lds_off(const void* p) {
  return (uint32_t)(uintptr_t)p;
}

// Adjacent-sample pair fetched with one (possibly 4B-aligned) b64 load.
struct __align__(4) FPair {
  float a, b;
};
__device__ __forceinline__ FPair load_pair(const float* p) {
  return *reinterpret_cast<const FPair*>(p);
}

__global__ __launch_bounds__(NT) void stereo_loss_tile_kernel(
    const float* __restrict__ left, const float* __restrict__ right,
    const float* __restrict__ dL, const float* __restrict__ dR,
    const float* __restrict__ non, float* __restrict__ partial) {
  __shared__ __align__(16) float l_lds[kC * HALO_H * ROWF];  // left tile + halo
  __shared__ __align__(16) float w_lds[kC * HALO_H * ROWF];  // warped right
  __shared__ __align__(16) float d_lds[HALO_H * ROWF];       // d_left tile
  __shared__ float red[3 * NT];

  const int tid = threadIdx.x;
  const int tx = blockIdx.x, ty = blockIdx.y, b = blockIdx.z;
  const int x0 = tx * TILE_W, y0 = ty * TILE_H;

  const float* limg = left + (size_t)b * kC * kHW;
  const float* rimg = right + (size_t)b * kC * kHW;
  const float* dl = dL + (size_t)b * kHW;
  const float* dr = dR + (size_t)b * kHW;
  const float* nw = non + (size_t)b * kHW;

  // ---- Phase A: async-stage left (3ch) and d_left halo tiles into LDS ----
  // Rows are padded to 72 floats; every 16B chunk is entirely inside or
  // outside the image (x0 multiple of 64, W = 960), so out-of-image chunks
  // are zero-filled via DS stores (count_include_pad=False semantics).
  constexpr int kTotalChunks = (kC + 1) * HALO_H * NCHUNK;  // 72 rows * 18
  for (int i = tid; i < kTotalChunks; i += NT) {
    const int ck = i % NCHUNK;
    const int row = i / NCHUNK;  // [0, 54): left channels; [54, 72): d_left
    const int col4 = ck * 4;
    const int gxs = x0 - 4 + col4;

    float* dst;
    const float* srcplane;
    int ry;
    if (row < kC * HALO_H) {
      const int c = row / HALO_H;
      ry = row % HALO_H;
      dst = &l_lds[(c * HALO_H + ry) * ROWF + col4];
      srcplane = limg + (size_t)c * kHW;
    } else {
      ry = row - kC * HALO_H;
      dst = &d_lds[ry * ROWF + col4];
      srcplane = dl;
    }
    const int gy = y0 - 1 + ry;
    if ((unsigned)gy < (unsigned)kH && gxs >= 0 && gxs <= kW - 4) {
      async_load_b128(lds_off(dst), srcplane + (size_t)gy * kW + (size_t)gxs);
    } else {
      *(float4*)dst = make_float4(0.f, 0.f, 0.f, 0.f);
    }
  }
  wait_async_all();
  __syncthreads();

  // ---- Phase B: horizontal bilinear warp of right image over the halo ----
  for (int i = tid; i < HALO_H * HALO_W; i += NT) {
    const int ry = i / HALO_W;
    const int hx = i % HALO_W;
    const int gy = y0 - 1 + ry;
    const int gx = x0 - 1 + hx;
    const int col = hx + 3;  // LDS col for global x = x0 - 4 + col
    float w0 = 0.f, w1 = 0.f, w2 = 0.f;
    if ((unsigned)gy < (unsigned)kH && (unsigned)gx < (unsigned)kW) {
      const float d = d_lds[ry * ROWF + col];
      const float xs = (float)gx - d;
      const float xf = floorf(xs);
      const float fr = xs - xf;
      const float ifr = 1.f - fr;
      const int xi = (int)xf;
      const int xc = min(max(xi, 0), kW - 2);
      const bool lo = (xi < 0);        // both taps clamp to column 0
      const bool hi = (xi >= kW - 1);  // both taps clamp to column W-1
      const float* r0 = rimg + (size_t)gy * kW;
      {
        const FPair p = load_pair(r0 + xc);
        const float v0 = hi ? p.b : p.a;
        const float v1 = lo ? p.a : p.b;
        w0 = ifr * v0 + fr * v1;
      }
      {
        const FPair p = load_pair(r0 + kHW + xc);
        const float v0 = hi ? p.b : p.a;
        const float v1 = lo ? p.a : p.b;
        w1 = ifr * v0 + fr * v1;
      }
      {
        const FPair p = load_pair(r0 + 2 * kHW + xc);
        const float v0 = hi ? p.b : p.a;
        const float v1 = lo ? p.a : p.b;
        w2 = ifr * v0 + fr * v1;
      }
    }
    w_lds[(0 * HALO_H + ry) * ROWF + col] = w0;
    w_lds[(1 * HALO_H + ry) * ROWF + col] = w1;
    w_lds[(2 * HALO_H + ry) * ROWF + col] = w2;
  }
  __syncthreads();

  // ---- Phase C: fused SSIM + Charbonnier + LR terms, per-thread accumulate --
  float acc_pv = 0.f, acc_v = 0.f, acc_lr = 0.f;
#pragma unroll
  for (int k = 0; k < (TILE_H * TILE_W) / NT; ++k) {
    const int op = tid + k * NT;
    const int oy = op / TILE_W, ox = op % TILE_W;
    const int gy = y0 + oy, gx = x0 + ox;

    // avg_pool3 count_include_pad=False: analytic valid-window count.
    const float cy = 1.f + (gy > 0 ? 1.f : 0.f) + (gy < kH - 1 ? 1.f : 0.f);
    const float cx = 1.f + (gx > 0 ? 1.f : 0.f) + (gx < kW - 1 ? 1.f : 0.f);
    const float inv = 1.f / (cy * cx);

    float s_sum = 0.f, l1_sum = 0.f;
#pragma unroll
    for (int c = 0; c < kC; ++c) {
      const float* lb = &l_lds[(c * HALO_H + oy) * ROWF + ox + 3];
      const float* wb = &w_lds[(c * HALO_H + oy) * ROWF + ox + 3];
      float Sx = 0.f, Sy = 0.f, Sxx = 0.f, Syy = 0.f, Sxy = 0.f;
#pragma unroll
      for (int dy = 0; dy < 3; ++dy) {
#pragma unroll
        for (int dx = 0; dx < 3; ++dx) {
          const float lv = lb[dy * ROWF + dx];
          const float wv = wb[dy * ROWF + dx];
          Sx += lv;
          Sy += wv;
          Sxx = fmaf(lv, lv, Sxx);
          Syy = fmaf(wv, wv, Syy);
          Sxy = fmaf(lv, wv, Sxy);
        }
      }
      const float mux = Sx * inv, muy = Sy * inv;
      const float sgx = Sxx * inv - mux * mux;
      const float sgy = Syy * inv - muy * muy;
      const float sgxy = Sxy * inv - mux * muy;
      const float num = (2.f * mux * muy + kC1) * (2.f * sgxy + kC2);
      const float den = (mux * mux + muy * muy + kC1) * (sgx + sgy + kC2);
      const float ssim = num / (den + 1e-12f);
      s_sum += fminf(fmaxf((1.f - ssim) * 0.5f, 0.f), 1.f);

      const float lcv = lb[ROWF + 1];  // center (dy=1, dx=1)
      const float wcv = wb[ROWF + 1];
      const float df = lcv - wcv;
      l1_sum += sqrtf(fmaf(df, df, kEps2));
    }
    const float photo =
        kAlpha * (s_sum * (1.f / 3.f)) + (1.f - kAlpha) * (l1_sum * (1.f / 3.f));

    // validity mask (valid_r == valid mask: same disparity drives both warps)
    const float dv = d_lds[(oy + 1) * ROWF + ox + 4];
    const float xs = (float)gx - dv;
    const float mask = (xs >= 0.f && xs <= (float)(kW - 1)) ? 1.f : 0.f;
    const float valid = mask * nw[(size_t)gy * kW + gx];

    // LR-consistency term: warp d_right by d_left at the center pixel.
    const float xf = floorf(xs);
    const float fr = xs - xf;
    const int xi = (int)xf;
    const int xc = min(max(xi, 0), kW - 2);
    const bool lolim = (xi < 0);
    const bool hilim = (xi >= kW - 1);
    const FPair p = load_pair(dr + (size_t)gy * kW + xc);
    const float v0 = hilim ? p.b : p.a;
    const float v1 = lolim ? p.a : p.b;
    const float dw = (1.f - fr) * v0 + fr * v1;
    const float dd = dv - dw;
    const float lr = sqrtf(fmaf(dd, dd, kEps2));

    acc_pv = fmaf(photo, valid, acc_pv);
    acc_v += valid;
    acc_lr = fmaf(lr, valid, acc_lr);
  }

  // ---- Deterministic block reduction ----
  red[tid] = acc_pv;
  red[NT + tid] = acc_v;
  red[2 * NT + tid] = acc_lr;
  __syncthreads();
#pragma unroll
  for (int s = NT / 2; s > 0; s >>= 1) {
    if (tid < s) {
      red[tid] += red[tid + s];
      red[NT + tid] += red[NT + tid + s];
      red[2 * NT + tid] += red[2 * NT + tid + s];
    }
    __syncthreads();
  }
  if (tid == 0) {
    const int blin = (b * TILES_Y + ty) * TILES_X + tx;
    partial[3 * blin + 0] = red[0];
    partial[3 * blin + 1] = red[NT];
    partial[3 * blin + 2] = red[2 * NT];
  }
}

__global__ __launch_bounds__(NT) void stereo_loss_finalize_kernel(
    const float* __restrict__ partial, int nblocks, float* __restrict__ out,
    float inv_count) {
  __shared__ float red[3 * NT];
  const int tid = threadIdx.x;
  float a0 = 0.f, a1 = 0.f, a2 = 0.f;
  for (int i = tid; i < nblocks; i += NT) {
    a0 += partial[3 * i + 0];
    a1 += partial[3 * i + 1];
    a2 += partial[3 * i + 2];
  }
  red[tid] = a0;
  red[NT + tid] = a1;
  red[2 * NT + tid] = a2;
  __syncthreads();
#pragma unroll
  for (int s = NT / 2; s > 0; s >>= 1) {
    if (tid < s) {
      red[tid] += red[tid + s];
      red[NT + tid] += red[NT + tid + s];
      red[2 * NT + tid] += red[2 * NT + tid + s];
    }
    __syncthreads();
  }
  if (tid == 0) {
    const float s_pv = red[0];
    const float s_v = red[NT];
    const float s_lr = red[2 * NT];
    const float photo_loss = s_pv / (s_v + 1e-6f);
    const float lr_loss = s_lr / (s_v + 1e-6f);
    out[0] = photo_loss + kWlr * lr_loss;  // total (W_PHOTO = 1)
    out[1] = photo_loss;
    out[2] = lr_loss;
    out[3] = s_v * inv_count;  // valid_ratio = mean(valid)
  }
}

extern "C" void kernel_launch(void* const* d_in, const int* in_sizes, int n_in,
                              void* d_out, int out_size, void* d_ws,
                              size_t ws_size, hipStream_t stream) {
  (void)n_in;
  (void)out_size;
  (void)ws_size;
  const float* left = (const float*)d_in[0];
  const float* right = (const float*)d_in[1];
  const float* dL = (const float*)d_in[2];
  const float* dR = (const float*)d_in[3];
  const float* non = (const float*)d_in[4];

  const int B = in_sizes[0] / (kC * kH * kW);  // 8 for the reference shapes
  float* partial = (float*)d_ws;               // 3 floats per tile block

  dim3 grid(TILES_X, TILES_Y, B);
  hipLaunchKernelGGL(stereo_loss_tile_kernel, grid, dim3(NT), 0, stream, left,
                     right, dL, dR, non, partial);

  const int nblocks = TILES_X * TILES_Y * B;
  hipLaunchKernelGGL(stereo_loss_finalize_kernel, dim3(1), dim3(NT), 0, stream,
                     partial, nblocks, (float*)d_out,
                     1.0f / (float)(B * kH * kW));
}